// HausdorffDTLoss_13434657702678
// MI455X (gfx1250) — compile-verified
//
#include <hip/hip_runtime.h>
#include <hip/hip_bf16.h>

// Problem constants for (2,1,128,128,128) volumes.
#define NB_DIM   2
#define NAX      128
#define VOL      (128 * 128 * 128)            // per-sample voxels = 2,097,152
#define TOT      (NB_DIM * VOL)               // total voxels      = 4,194,304
#define NLINES   (TOT / NAX)                  // lines per pass    = 32,768
#define BIGF     49153.0f                     // 3*128^2 + 1, > any achievable sq dist
#define RED_NB   1024                         // reduction blocks
#define RED_NT   256                          // reduction threads/block

typedef __attribute__((ext_vector_type(2))) float v2f;
typedef __attribute__((ext_vector_type(8))) float v8f;

// ---------------------------------------------------------------------------
// Zero the accumulated squared-distance field and the fg_any flags.
__global__ void hdt_zero_kernel(float* __restrict__ dist, int* __restrict__ flags) {
    int idx = blockIdx.x * blockDim.x + threadIdx.x;
    if (idx < TOT) dist[idx] = 0.0f;
    if (idx < 4)   flags[idx] = 0;
}

// ---------------------------------------------------------------------------
// Build the two sqEDT seed channels for one image:
//   ch0 (work[0..TOT))      : f = fg ? BIG : 0   -> sq dist (at fg voxels) to nearest bg
//   ch1 (work[TOT..2*TOT))  : f = fg ? 0 : BIG   -> sq dist (at bg voxels) to nearest fg
// Also record per-sample fg_any via one atomic per wave.
__global__ void hdt_init_kernel(const float* __restrict__ img,
                                float* __restrict__ work,
                                int* __restrict__ flags) {
    int idx = blockIdx.x * blockDim.x + threadIdx.x;
    bool fg = img[idx] > 0.5f;
    work[idx]       = fg ? BIGF : 0.0f;
    work[TOT + idx] = fg ? 0.0f : BIGF;
    // VOL is a multiple of the wave size, so `sample` is wave-uniform.
    if (__any(fg) && (threadIdx.x & 31) == 0) {
        atomicOr(&flags[idx / VOL], 1);
    }
}

// ---------------------------------------------------------------------------
// One in-place 1D squared-EDT pass along a given axis.
//   g[i] = min_j f[j] + (i-j)^2  =  i^2 + min_j ( (f[j] + j^2) - 2*i*j )
// blockIdx.x = line index, blockIdx.y = channel (0: fg-EDT, 1: bg-EDT).
// 128 threads/block. The raw line is staged memory->LDS with the gfx1250
// async-to-LDS path (no VGPR bounce, tracked on ASYNCcnt), then biased by
// +j^2 in place. The scan is a fully unrolled 128-deep fma+min over LDS
// broadcast reads; the final +i^2 is folded into the writeback.
__global__ void hdt_edt_pass_kernel(float* __restrict__ work,
                                    int outer_mult, int l_inner, int inner_mult,
                                    int stride) {
    __shared__ float h[NAX];
    float* f = work + (size_t)blockIdx.y * TOT;
    const int l    = blockIdx.x;
    const int base = (l / l_inner) * outer_mult + (l % l_inner) * inner_mult;
    const int i    = threadIdx.x;               // output index along the line
    const int gi   = base + i * stride;

    // ---- async stage: copy f[gi] from global memory into h[i] in LDS  (GLOBAL_LOAD_ASYNC_TO_LDS_B32) ----
    {
        const float* gp = f + gi;                               // 64-bit VGPR pair
        unsigned lds_off = (unsigned)(uintptr_t)(&h[i]);        // low 32 bits = LDS offset
        asm volatile("global_load_async_to_lds_b32 %0, %1, off"
                     :
                     : "v"(lds_off), "v"(gp)
                     : "memory");
        asm volatile("s_wait_asynccnt 0x0" ::: "memory");       // my wave's DMA done
    }
    __syncthreads();                                            // all waves' lines visible

    const float fi = (float)i;
    const float v  = h[i];                       // raw f[i] (own-element read for in-place)
    h[i] = fmaf(fi, fi, v);                      // f[i] + i^2
    __syncthreads();

    const float n2i = -2.0f * fi;
    float m = 3.0e38f;
#pragma unroll
    for (int j = 0; j < NAX; ++j) {
        m = fminf(m, fmaf(n2i, (float)j, h[j]));
    }
    f[gi] = fmaf(fi, fi, m);                     // + i^2 ; in-place (own element only)
}

// ---------------------------------------------------------------------------
// dist += (ch0 + ch1) * fg_any   (per-sample guard, matching the reference)
__global__ void hdt_accum_kernel(const float* __restrict__ work,
                                 float* __restrict__ dist,
                                 const int* __restrict__ flags, int slot) {
    int idx = blockIdx.x * blockDim.x + threadIdx.x;
    float g = flags[slot + idx / VOL] ? 1.0f : 0.0f;
    dist[idx] += (work[idx] + work[TOT + idx]) * g;
}

// ---------------------------------------------------------------------------
// Stage-1 reduction of sum( (p-t)^2 * dist ) using the f32 matrix pipe:
// each lane feeds its product into A(K=0) of V_WMMA_F32_16X16X4_F32 with
// B = ones, accumulating into a 16x16 f32 tile whose grand total is
// 16 x (true sum). Grid is sized so TOT % nthreads == 0 -> every lane runs
// the same trip count -> EXEC is all-ones at every WMMA (ISA requirement).
__global__ void hdt_reduce_kernel(const float* __restrict__ pred,
                                  const float* __restrict__ target,
                                  const float* __restrict__ dist,
                                  float* __restrict__ partial) {
    const int tid      = blockIdx.x * blockDim.x + threadIdx.x;
    const int nthreads = RED_NB * RED_NT;       // TOT / nthreads == 16 exactly

    v8f acc = {};
    v2f b;  b[0] = 1.0f;  b[1] = 1.0f;

    for (int idx = tid; idx < TOT; idx += nthreads) {
        __builtin_prefetch(&pred[idx + nthreads], 0, 0);
        __builtin_prefetch(&dist[idx + nthreads], 0, 0);
        float d = pred[idx] - target[idx];
        float q = d * d * dist[idx];
        v2f a;  a[0] = q;  a[1] = 0.0f;
        // D = A(16x4) x B(4x16) + C : per step adds lane products into the tile.
        acc = __builtin_amdgcn_wmma_f32_16x16x4_f32(
            /*neg_a=*/false, a, /*neg_b=*/false, b,
            /*c_mod=*/(short)0, acc, /*reuse_a=*/false, /*reuse_b=*/false);
    }

    float s = acc[0] + acc[1] + acc[2] + acc[3] + acc[4] + acc[5] + acc[6] + acc[7];

    __shared__ float sm[RED_NT];
    sm[threadIdx.x] = s;
    __syncthreads();
    for (int off = RED_NT / 2; off > 0; off >>= 1) {
        if (threadIdx.x < off) sm[threadIdx.x] += sm[threadIdx.x + off];
        __syncthreads();
    }
    if (threadIdx.x == 0) partial[blockIdx.x] = sm[0];   // still 16x the true sum
}

// ---------------------------------------------------------------------------
// Stage-2: fold block partials, undo the 16x WMMA-column replication, take mean.
__global__ void hdt_final_kernel(const float* __restrict__ partial,
                                 float* __restrict__ out) {
    __shared__ float sm[RED_NT];
    float s = 0.0f;
    for (int i = threadIdx.x; i < RED_NB; i += RED_NT) s += partial[i];
    sm[threadIdx.x] = s;
    __syncthreads();
    for (int off = RED_NT / 2; off > 0; off >>= 1) {
        if (threadIdx.x < off) sm[threadIdx.x] += sm[threadIdx.x + off];
        __syncthreads();
    }
    if (threadIdx.x == 0) out[0] = sm[0] / (16.0f * (float)TOT);
}

// ---------------------------------------------------------------------------
extern "C" void kernel_launch(void* const* d_in, const int* in_sizes, int n_in,
                              void* d_out, int out_size, void* d_ws, size_t ws_size,
                              hipStream_t stream) {
    const float* pred   = (const float*)d_in[0];
    const float* target = (const float*)d_in[1];
    float* out = (float*)d_out;

    // Workspace layout (floats): dist[TOT] | work[2*TOT] | partial[RED_NB] | flags[4 ints]
    float* dist    = (float*)d_ws;
    float* work    = dist + TOT;
    float* partial = work + (size_t)2 * TOT;
    int*   flags   = (int*)(partial + RED_NB);

    const int NT = 256;
    const int NBLK = TOT / NT;
    dim3 edt_grid(NLINES, 2);   // lines x channels

    hdt_zero_kernel<<<NBLK, NT, 0, stream>>>(dist, flags);

    // ---- pred image: seed, 3 separable passes (Z, Y, X), accumulate ----
    hdt_init_kernel<<<NBLK, NT, 0, stream>>>(pred, work, flags + 0);
    hdt_edt_pass_kernel<<<edt_grid, NAX, 0, stream>>>(work, NAX,        1,     0, 1);      // Z (contig)
    hdt_edt_pass_kernel<<<edt_grid, NAX, 0, stream>>>(work, NAX * NAX,  NAX,   1, NAX);    // Y
    hdt_edt_pass_kernel<<<edt_grid, NAX, 0, stream>>>(work, VOL,  NAX * NAX,   1, NAX*NAX);// X
    hdt_accum_kernel<<<NBLK, NT, 0, stream>>>(work, dist, flags, 0);

    // ---- target image ----
    hdt_init_kernel<<<NBLK, NT, 0, stream>>>(target, work, flags + 2);
    hdt_edt_pass_kernel<<<edt_grid, NAX, 0, stream>>>(work, NAX,        1,     0, 1);
    hdt_edt_pass_kernel<<<edt_grid, NAX, 0, stream>>>(work, NAX * NAX,  NAX,   1, NAX);
    hdt_edt_pass_kernel<<<edt_grid, NAX, 0, stream>>>(work, VOL,  NAX * NAX,   1, NAX*NAX);
    hdt_accum_kernel<<<NBLK, NT, 0, stream>>>(work, dist, flags, 2);

    // ---- loss = mean( (p-t)^2 * dist ) via f32 WMMA reduction ----
    hdt_reduce_kernel<<<RED_NB, RED_NT, 0, stream>>>(pred, target, dist, partial);
    hdt_final_kernel<<<1, RED_NT, 0, stream>>>(partial, out);
}